// ChirpLinker_87651692576915
// MI455X (gfx1250) — compile-verified
//
#include <hip/hip_runtime.h>
#include <math.h>
#include <stdint.h>

#define B_ 128
#define W_ 64
#define K_ 32
#define F_ 9
#define C_ (W_ * K_)            // 2048 candidates
#define SLAB_ (W_ * K_ * F_)    // 18432 floats = 73728 bytes per batch

// LDS layout offsets (bytes)
#define OFF_TOK    0            // 18432 floats  -> 73728
#define OFF_BEST   73728        // 2048 floats   -> 8192
#define OFF_PAR    81920        // 2048 ints     -> 8192
#define OFF_KEYS   90112        // 2048 u64      -> 16384 (8B aligned)
#define OFF_CID    106496       // 2048 ints     -> 8192
#define OFF_EXW    114688       // 64 u32        -> 256
#define OFF_USED   114944       // 64 u32        -> 256
#define OFF_CHAIN  115200       // 64 ints       -> 256
#define SMEM_BYTES 115456

typedef int i32x8 __attribute__((ext_vector_type(8)));
typedef int i32x4 __attribute__((ext_vector_type(4)));

#if defined(__AMDGCN__) && __has_builtin(__builtin_amdgcn_tensor_load_to_lds) && \
    __has_builtin(__builtin_amdgcn_tensor_store_from_lds)
#define HAVE_TDM 1
#else
#define HAVE_TDM 0
#endif

__device__ __forceinline__ float wrapf(float x) {
  const float PI_ = 3.14159265358979323846f;
  const float TWO_PI_ = 6.2831853071795864769f;
  float m = fmodf(x + PI_, TWO_PI_);
  if (m < 0.f) m += TWO_PI_;      // Python-style mod: result in [0, 2pi)
  return m - PI_;
}

#if HAVE_TDM
// Build the D# for a flat 1-row 2-D tensor of SLAB_ 4-byte elements.
__device__ __forceinline__ void tdm_make_desc(const void* gptr, unsigned int lds_off,
                                              i32x4& g0, i32x8& g1) {
  unsigned long long ga = (unsigned long long)(uintptr_t)gptr;
  // group 0: count=1 | lds_addr | global_addr[56:0] | type=2
  g0[0] = 1;
  g0[1] = (int)lds_off;
  g0[2] = (int)(unsigned int)ga;
  g0[3] = (int)(((unsigned int)(ga >> 32) & 0x01FFFFFFu) | 0x80000000u);
  // group 1: data_size=4B; tensor_dim0 = tile_dim0 = stride0 = 18432; dim1 = tile1 = 1
  g1[0] = 0x00020000;        // workgroup_mask=0, data_size=2 (4B)
  g1[1] = (int)0x48000000;   // tensor_dim0 = 18432 (low 16 in bits 63:48)
  g1[2] = 0x00010000;        // tensor_dim0 hi=0, tensor_dim1 = 1
  g1[3] = (int)0x48000000;   // tensor_dim1 hi=0, tile_dim0 = 18432
  g1[4] = 1;                 // tile_dim1 = 1, tile_dim2 = 0
  g1[5] = 18432;             // tensor_dim0_stride (lo 32)
  g1[6] = 0;                 // stride hi / dim1_stride lo
  g1[7] = 0;
}
#endif

__global__ __launch_bounds__(32) void ChirpLinker_main(
    const float* __restrict__ tokens, float* __restrict__ out,
    int* __restrict__ nch) {
  extern __shared__ unsigned char smem[];
  float* tok  = (float*)(smem + OFF_TOK);
  float* bestA = (float*)(smem + OFF_BEST);
  int*   par   = (int*)(smem + OFF_PAR);
  unsigned long long* keys = (unsigned long long*)(smem + OFF_KEYS);
  int*   cidA  = (int*)(smem + OFF_CID);
  unsigned int* exw  = (unsigned int*)(smem + OFF_EXW);
  unsigned int* used = (unsigned int*)(smem + OFF_USED);
  int* chainTok = (int*)(smem + OFF_CHAIN);

  const int tid = threadIdx.x;          // lane == k state, wave32
  const int b = blockIdx.x;
  const float* gtok = tokens + (size_t)b * SLAB_;

  // -------- Phase 0: TDM bulk DMA of this batch's token slab into LDS ------
#if HAVE_TDM
  {
    i32x4 g0; i32x8 g1;
    tdm_make_desc(gtok, (unsigned int)(uintptr_t)tok, g0, g1);
    i32x4 z4 = {0, 0, 0, 0};
    i32x8 z8 = {0, 0, 0, 0, 0, 0, 0, 0};
    __builtin_amdgcn_tensor_load_to_lds(g0, g1, z4, z4, z8, 0);
    __builtin_amdgcn_s_wait_tensorcnt(0);
  }
#else
  for (int i = tid; i < SLAB_; i += 32) tok[i] = gtok[i];
#endif
  __syncthreads();

  // -------- Phase 1: DP over windows; lane = k_next, shuffles for k_prev ---
  float s0 = tok[(0 * K_ + tid) * F_ + 0];
  bool ex = s0 > 0.f;
  float best = ex ? s0 * s0 : -INFINITY;
  bestA[tid] = best;
  par[tid] = -1;
  {
    unsigned long long m = __ballot(ex);
    if (tid == 0) exw[0] = (unsigned int)m;
  }
  for (int w = 1; w < W_; ++w) {
    const float* nf = &tok[(w * K_ + tid) * F_];
    float snrN = nf[0], fS = nf[3], aS = nf[5], ps = nf[7];
    const float* pf = &tok[((w - 1) * K_ + tid) * F_];
    float snrP_m = pf[0], fE_m = pf[4], aE_m = pf[6], pe_m = pf[8];
    float bprev = best;
    int exprev = ex ? 1 : 0;
    float bm = -INFINITY;
    int pmax = -1;
    for (int kp = 0; kp < K_; ++kp) {
      float bP  = __shfl(bprev, kp, 32);
      int   eP  = __shfl(exprev, kp, 32);
      float snrP = __shfl(snrP_m, kp, 32);
      float fE  = __shfl(fE_m, kp, 32);
      float aE  = __shfl(aE_m, kp, 32);
      float pe  = __shfl(pe_m, kp, 32);
      bool act = (snrP > 0.f) && (snrN > 0.f);
      float fm = 0.5f * (fE + fS);
      bool df_ok = !((fm > 0.f) && (fabsf(fE - fS) > 0.05f * fm));
      bool dphi_ok = fabsf(wrapf(ps - pe)) <= 0.5f;
      float am = fmaxf(aE, aS);
      bool dA_ok = !((am > 0.f) && (fabsf(aE - aS) > 0.5f * am));
      bool edge = act && df_ok && dphi_ok && dA_ok;
      float sc = (edge && eP) ? bP : -INFINITY;
      if (sc > bm) { bm = sc; pmax = kp; }   // first-max tie-break (argmax)
    }
    bool exn = pmax >= 0;
    best = exn ? (bm + snrN * snrN) : -INFINITY;
    ex = exn;
    bestA[w * K_ + tid] = best;
    par[w * K_ + tid] = exn ? pmax : -1;
    unsigned long long m = __ballot(exn);
    if (tid == 0) exw[w] = (unsigned int)m;
  }
  __syncthreads();

  // -------- Phase 2: candidate scores -> sortable 64-bit keys --------------
  for (int c = tid; c < C_; c += 32) {
    int w = c >> 5, k = c & 31;
    float s = tok[c * F_ + 0];
    bool candb = s > 0.f;
    bool exb = (exw[w] >> k) & 1u;
    float t = candb ? (exb ? bestA[c] : s * s) : -INFINITY;
    unsigned int u = __float_as_uint(t);
    u = (u & 0x80000000u) ? ~u : (u | 0x80000000u); // ascending-order map
    unsigned int hi = ~u;                            // descending score
    keys[c] = ((unsigned long long)hi << 32) | (unsigned int)c;
    cidA[c] = -1;
  }
  for (int w = tid; w < W_; w += 32) used[w] = 0u;
  __syncthreads();

  // -------- Phase 3: bitonic sort of 2048 keys (ascending) -----------------
  for (int ksz = 2; ksz <= C_; ksz <<= 1) {
    for (int j = ksz >> 1; j > 0; j >>= 1) {
      for (int i = tid; i < C_; i += 32) {
        int ixj = i ^ j;
        if (ixj > i) {
          unsigned long long a = keys[i], c2 = keys[ixj];
          bool up = ((i & ksz) == 0);
          if ((a > c2) == up) { keys[i] = c2; keys[ixj] = a; }
        }
      }
      __syncthreads();
    }
  }

  // -------- Phase 4: sequential greedy selection + in-place apply ----------
  int cnt = 0;
  for (int oi = 0; oi < C_; ++oi) {
    int c = (int)(keys[oi] & 0xFFFFFFFFull);
    int we_ = c >> 5, ke_ = c & 31;
    float snrEnd = tok[(we_ * K_ + ke_) * F_ + 0];
    bool candf = snrEnd > 0.f;
    bool exEnd = (exw[we_] >> ke_) & 1u;
    int span = exEnd ? (we_ + 1) : 1;       // == length (sv.sum) in reference
    if (tid == 0) {                          // backtrack (sequential)
      int cur = ke_;
      for (int j = 0; j < span; ++j) {
        int t = cur < 0 ? 0 : cur;           // clip like reference
        chainTok[j] = t;
        cur = par[(we_ - j) * K_ + t];
      }
    }
    __syncthreads();
    int conf = 0;
    for (int j = tid; j < span; j += 32) {
      int w = we_ - j, t = chainTok[j];
      if ((used[w] >> t) & 1u) conf = 1;
    }
    bool conflict = __any(conf) != 0;
    bool acc = candf && !conflict;
    if (acc) {
      for (int j = tid; j < span; j += 32) {
        int w = we_ - j, t = chainTok[j];
        used[w] |= (1u << t);                // slots within a chain are unique
      }
      if (span >= 2) {                       // MIN_LENGTH
        // combined SNR over chain slots (reads before writes)
        float psum = 0.f;
        for (int j = tid; j < span; j += 32) {
          int w = we_ - j, t = chainTok[j];
          float s = tok[(w * K_ + t) * F_ + 0];
          psum += s * s;
        }
        for (int off = 16; off; off >>= 1) psum += __shfl_xor(psum, off, 32);
        float comb = sqrtf(psum);
        __syncthreads();
        for (int j = tid; j < span; j += 32) {
          int w = we_ - j, t = chainTok[j];
          tok[(w * K_ + t) * F_ + 0] = comb;   // s + (comb - s)
          cidA[w * K_ + t] = cnt;
        }
        // junction smoothing: pair p = (slot p [later], slot p+1 [earlier]);
        // each pair's read/write footprint (slot,col) is disjoint from others
        for (int p = tid; p < span - 1; p += 32) {
          int wn = we_ - p,       tn = chainTok[p];
          int wp = we_ - (p + 1), tp = chainTok[p + 1];
          float* P = &tok[(wp * K_ + tp) * F_];
          float* N = &tok[(wn * K_ + tn) * F_];
          float fe = P[4], fs = N[3];
          float ae = P[6], as_ = N[5];
          float corr = wrapf(N[7] - P[8]);
          float favg = 0.5f * (fe + fs);
          float aavg = 0.5f * (ae + as_);
          P[4] = favg; N[3] = favg;
          P[6] = aavg; N[5] = aavg;
          P[8] = P[8] + 0.5f * corr;
          N[7] = N[7] - 0.5f * corr;
        }
        ++cnt;                                // uniform across lanes
      }
    }
    __syncthreads();
  }

  // -------- Phase 5: flush results ----------------------------------------
  float* dataOut = out + (size_t)b * SLAB_;
#if HAVE_TDM
  {
    // LDS tok slab now holds the final data; DMA it back with the TDM.
    __syncthreads();                         // all LDS writes visible
    i32x4 g0; i32x8 g1;
    tdm_make_desc(dataOut, (unsigned int)(uintptr_t)tok, g0, g1);
    i32x4 z4 = {0, 0, 0, 0};
    i32x8 z8 = {0, 0, 0, 0, 0, 0, 0, 0};
    __builtin_amdgcn_tensor_store_from_lds(g0, g1, z4, z4, z8, 0);
    // completion enforced before wave end (S_ENDPGM implies wait-idle),
    // but wait explicitly so the cid stores below can't be reordered around it
    __builtin_amdgcn_s_wait_tensorcnt(0);
  }
#else
  for (int i = tid; i < SLAB_; i += 32) dataOut[i] = tok[i];
#endif
  int* cidOut = (int*)(out + (size_t)B_ * SLAB_) + (size_t)b * C_;
  for (int i = tid; i < C_; i += 32) cidOut[i] = cidA[i];
  if (tid == 0) nch[b] = cnt;
}

// Cross-batch chain-id offset fixup: cid += exclusive_cumsum(n_chains)[b]
__global__ void ChirpFinalize(int* __restrict__ cid, const int* __restrict__ nch,
                              int total) {
  __shared__ int offs[B_];
  if (threadIdx.x == 0) {
    int acc = 0;
    for (int i = 0; i < B_; ++i) { offs[i] = acc; acc += nch[i]; }
  }
  __syncthreads();
  int stride = gridDim.x * blockDim.x;
  for (int i = blockIdx.x * blockDim.x + threadIdx.x; i < total; i += stride) {
    int v = cid[i];
    if (v >= 0) cid[i] = v + offs[i / C_];
  }
}

extern "C" void kernel_launch(void* const* d_in, const int* in_sizes, int n_in,
                              void* d_out, int out_size, void* d_ws, size_t ws_size,
                              hipStream_t stream) {
  (void)in_sizes; (void)n_in; (void)out_size; (void)ws_size;
  const float* tokens = (const float*)d_in[0];
  float* out = (float*)d_out;
  int* nch = (int*)d_ws;                 // 128 ints of scratch

  ChirpLinker_main<<<B_, 32, SMEM_BYTES, stream>>>(tokens, out, nch);

  int* cid = (int*)(out + (size_t)B_ * SLAB_);
  ChirpFinalize<<<64, 256, 0, stream>>>(cid, nch, B_ * C_);
}